// SimGCN_29386166239469
// MI455X (gfx1250) — compile-verified
//
#include <hip/hip_runtime.h>
#include <math.h>

// SimGCN fused pipeline for gfx1250 (MI455X), wave32 + V_WMMA_F32_16X16X4_F32.
//   K1: h = ((x@W1+b1)@W2+b2)@W3+b3           [12288,16]  (VALU, weights in LDS)
//   K2: flash-style  sim@h  without materializing the 12288^2 adj matrix:
//       per 16x16 tile: T = Hj @ Hi^T (4 chained f32 WMMAs), P = (T>0)?exp(T):0,
//       O += P^T-as-A @ Hj (4 WMMAs), running per-query sum l += sum_k P.
//       (softmax max-subtraction skipped: scores bounded ~O(few), shift-invariant)
//   K3: out = log_softmax(O / l) per row of 16.

#define NROWS 12288
#define NDIM  128
#define HID   16

typedef float v2f __attribute__((ext_vector_type(2)));
typedef float v8f __attribute__((ext_vector_type(8)));

__device__ __forceinline__ v8f wmma4(v2f a, v2f b, v8f c) {
  // D = A(16x4 f32) x B(4x16 f32) + C(16x16 f32)
  return __builtin_amdgcn_wmma_f32_16x16x4_f32(false, a, false, b, (short)0, c,
                                               false, false);
}

// ---------------- Kernel 1: 3-layer MLP ----------------
__global__ __launch_bounds__(256) void mlp16_kernel(
    const float* __restrict__ x,
    const float* __restrict__ W1, const float* __restrict__ b1,
    const float* __restrict__ W2, const float* __restrict__ b2,
    const float* __restrict__ W3, const float* __restrict__ b3,
    float* __restrict__ h) {
  __shared__ float sW1[NDIM * HID];
  __shared__ float sW2[HID * HID];
  __shared__ float sW3[HID * HID];
  __shared__ float sb1[HID], sb2[HID], sb3[HID];
  int tid = threadIdx.x;
  for (int i = tid; i < NDIM * HID; i += 256) sW1[i] = W1[i];
  if (tid < HID * HID) { sW2[tid] = W2[tid]; sW3[tid] = W3[tid]; }
  if (tid < HID) { sb1[tid] = b1[tid]; sb2[tid] = b2[tid]; sb3[tid] = b3[tid]; }
  __syncthreads();

  int row = blockIdx.x * 256 + tid;
  const float* xr = x + (size_t)row * NDIM;

  float a1[HID];
#pragma unroll
  for (int j = 0; j < HID; ++j) a1[j] = sb1[j];
  for (int k = 0; k < NDIM; ++k) {
    float xv = xr[k];
#pragma unroll
    for (int j = 0; j < HID; ++j) a1[j] = fmaf(xv, sW1[k * HID + j], a1[j]);
  }
  float a2[HID];
#pragma unroll
  for (int j = 0; j < HID; ++j) a2[j] = sb2[j];
#pragma unroll
  for (int k = 0; k < HID; ++k) {
#pragma unroll
    for (int j = 0; j < HID; ++j) a2[j] = fmaf(a1[k], sW2[k * HID + j], a2[j]);
  }
  float a3[HID];
#pragma unroll
  for (int j = 0; j < HID; ++j) a3[j] = sb3[j];
#pragma unroll
  for (int k = 0; k < HID; ++k) {
#pragma unroll
    for (int j = 0; j < HID; ++j) a3[j] = fmaf(a2[k], sW3[k * HID + j], a3[j]);
  }
  float* hp = h + (size_t)row * HID;
#pragma unroll
  for (int j = 0; j < HID; ++j) hp[j] = a3[j];
}

// ---------------- Kernel 2: fused masked-softmax aggregation ----------------
// Wave handles 2 query tiles of 16. Block = 8 waves = 256 queries. Grid = 48.
__global__ __launch_bounds__(256) void sim_attn_kernel(
    const float* __restrict__ h, float* __restrict__ Oacc,
    float* __restrict__ lsum_out) {
  int lane = threadIdx.x & 31;
  int wave = threadIdx.x >> 5;
  int r = lane & 15;       // row-within-tile / query / column index
  int half = lane >> 4;    // lane half selects K-subblock
  bool h0 = (half == 0);
  int qb = blockIdx.x * 256 + wave * 32;  // queries [qb, qb+32)

  // Score-GEMM B operand: B[k][n] = h[i+n][k]; lane holds h[i+r][4s+2*half+{0,1}]
  v2f Bi[2][4];
#pragma unroll
  for (int t = 0; t < 2; ++t) {
    const float* bp = h + (size_t)(qb + t * 16 + r) * HID + 2 * half;
#pragma unroll
    for (int s = 0; s < 4; ++s) Bi[t][s] = *(const v2f*)(bp + 4 * s);
  }

  v8f O[2];
  float ls[2] = {0.f, 0.f};
#pragma unroll
  for (int t = 0; t < 2; ++t)
#pragma unroll
    for (int v = 0; v < 8; ++v) O[t][v] = 0.f;

  for (int j = 0; j < NROWS; j += 16) {
    // Score-GEMM A operand: A[m][k] = h[j+m][k]; same per-lane layout as Bi.
    v2f Aj[4];
    const float* ap = h + (size_t)(j + r) * HID + 2 * half;
#pragma unroll
    for (int s = 0; s < 4; ++s) Aj[s] = *(const v2f*)(ap + 4 * s);
    // Value-GEMM B operand: B[kk][n] = h[j+kk][n]; lane: n=r, kk=4s+2*half+{0,1}
    v2f Bj[4];
#pragma unroll
    for (int s = 0; s < 4; ++s) {
      int rr = j + 4 * s + 2 * half;
      Bj[s][0] = h[(size_t)rr * HID + r];
      Bj[s][1] = h[(size_t)(rr + 1) * HID + r];
    }

#pragma unroll
    for (int t = 0; t < 2; ++t) {
      // T[key m][query n] = h_{j+m} . h_{i+n}   (C-layout: key = vgpr+8*half)
      v8f T = {0.f, 0.f, 0.f, 0.f, 0.f, 0.f, 0.f, 0.f};
#pragma unroll
      for (int s = 0; s < 4; ++s) T = wmma4(Aj[s], Bi[t][s], T);

      // mask + exp (no max-shift needed; scores are O(few))
      float P[8], SW[8];
      float part = 0.f;
#pragma unroll
      for (int v = 0; v < 8; ++v) {
        float xv = T[v];
        float e = __expf(xv);
        P[v] = xv > 0.f ? e : 0.f;
        part += P[v];
      }
      ls[t] += part;  // this lane-half covers its 8 keys; halves merged at end

      // P (C-layout, key=vgpr/half, query=lane) -> A-operand slices:
      // only cross-half swaps needed, no full transpose.
#pragma unroll
      for (int v = 0; v < 8; ++v) SW[v] = __shfl_xor(P[v], 16, 32);
      v2f a;
      a[0] = h0 ? P[0] : SW[2]; a[1] = h0 ? P[1] : SW[3];
      O[t] = wmma4(a, Bj[0], O[t]);
      a[0] = h0 ? P[4] : SW[6]; a[1] = h0 ? P[5] : SW[7];
      O[t] = wmma4(a, Bj[1], O[t]);
      a[0] = h0 ? SW[0] : P[2]; a[1] = h0 ? SW[1] : P[3];
      O[t] = wmma4(a, Bj[2], O[t]);
      a[0] = h0 ? SW[4] : P[6]; a[1] = h0 ? SW[5] : P[7];
      O[t] = wmma4(a, Bj[3], O[t]);
    }
  }

#pragma unroll
  for (int t = 0; t < 2; ++t) {
    float lt = ls[t] + __shfl_xor(ls[t], 16, 32);  // merge key halves
    if (h0) lsum_out[qb + t * 16 + r] = lt;
    // O C-layout: vgpr v -> query row (v + 8*half), lane -> feature r
#pragma unroll
    for (int v = 0; v < 8; ++v) {
      int row = qb + t * 16 + v + 8 * half;
      Oacc[(size_t)row * HID + r] = O[t][v];
    }
  }
}

// ---------------- Kernel 3: normalize + log_softmax (in place on d_out) -----
__global__ __launch_bounds__(256) void logsoftmax_kernel(
    float* __restrict__ out, const float* __restrict__ lsum) {
  int row = blockIdx.x * 256 + threadIdx.x;
  float inv = 1.0f / lsum[row];
  float* p = out + (size_t)row * HID;
  float v[HID];
#pragma unroll
  for (int j = 0; j < HID; ++j) v[j] = p[j] * inv;
  float m = v[0];
#pragma unroll
  for (int j = 1; j < HID; ++j) m = fmaxf(m, v[j]);
  float s = 0.f;
#pragma unroll
  for (int j = 0; j < HID; ++j) s += __expf(v[j] - m);
  float lse = m + __logf(s);
#pragma unroll
  for (int j = 0; j < HID; ++j) p[j] = v[j] - lse;
}

extern "C" void kernel_launch(void* const* d_in, const int* in_sizes, int n_in,
                              void* d_out, int out_size, void* d_ws,
                              size_t ws_size, hipStream_t stream) {
  const float* x  = (const float*)d_in[0];
  const float* W1 = (const float*)d_in[1];
  const float* b1 = (const float*)d_in[2];
  const float* W2 = (const float*)d_in[3];
  const float* b2 = (const float*)d_in[4];
  const float* W3 = (const float*)d_in[5];
  const float* b3 = (const float*)d_in[6];
  float* out = (float*)d_out;

  float* h = (float*)d_ws;                  // NROWS*16 floats
  float* l = h + (size_t)NROWS * HID;       // NROWS floats

  mlp16_kernel<<<NROWS / 256, 256, 0, stream>>>(x, W1, b1, W2, b2, W3, b3, h);
  sim_attn_kernel<<<NROWS / 256, 256, 0, stream>>>(h, out, l);
  logsoftmax_kernel<<<NROWS / 256, 256, 0, stream>>>(out, l);
}